// TopologicalDecoder_27779848471404
// MI455X (gfx1250) — compile-verified
//
#include <hip/hip_runtime.h>

typedef float v2f __attribute__((ext_vector_type(2)));
typedef float v8f __attribute__((ext_vector_type(8)));

#define BATCH 262144
#define NTILES (BATCH / 16)

// GEMM shapes (K-steps of 4, N-tiles of 16)
#define K1 38   // K = 152 = 8*16 chart + 8 bias-blend + 16 tex
#define N1 4    // N = 64
#define K2 16   // K = 64
#define N2 4    // N = 64
#define K3 16   // K = 64
#define N3 2    // N = 32

// const staging offsets (floats)
#define OFF_CEN  0    // centers        [8][16]
#define OFF_CSQ  128  // |center|^2     [8]
#define OFF_RWT  136  // router_w       [8][16]
#define OFF_RB   264  // router_b       [8]
#define OFF_TEXB 272  // tex_b          [64]
#define OFF_LNG  336  // ln_g           [64]
#define OFF_LNB  400  // ln_b           [64]
#define OFF_B1   464  // b1             [64]
#define OFF_B2   528  // b2             [32]
#define NCONST   560

__device__ __forceinline__ v8f wmma4(v2f a, v2f b, v8f c) {
  // V_WMMA_F32_16X16X4_F32: D = A(16x4,f32) * B(4x16,f32) + C(16x16,f32)
  return __builtin_amdgcn_wmma_f32_16x16x4_f32(
      /*neg_a=*/false, a, /*neg_b=*/false, b,
      /*c_mod=*/(short)0, c, /*reuse_a=*/false, /*reuse_b=*/false);
}

__device__ __forceinline__ float gelu_exact(float x) {
  return 0.5f * x * (1.0f + erff(x * 0.70710678118654752f));
}

// W1' [152][64]: rows 0..127 chart_w[c][h][l] at r=c*16+l; 128..135 chart_b[c][h];
// 136..151 tex_w[h][l] at r=136+l
__device__ __forceinline__ float w1prime(int r, int h,
    const float* __restrict__ chart_w, const float* __restrict__ chart_b,
    const float* __restrict__ tex_w) {
  if (r < 128) { int c = r >> 4, l = r & 15; return chart_w[(c * 64 + h) * 16 + l]; }
  if (r < 136) { return chart_b[(r - 128) * 64 + h]; }
  return tex_w[h * 16 + (r - 136)];
}

__global__ __launch_bounds__(128) void topo_decoder_kernel(
    const float* __restrict__ z_geo, const float* __restrict__ z_tex,
    const float* __restrict__ chart_w, const float* __restrict__ chart_b,
    const float* __restrict__ router_w, const float* __restrict__ router_b,
    const float* __restrict__ centers, const float* __restrict__ tex_w,
    const float* __restrict__ tex_b, const float* __restrict__ ln_g,
    const float* __restrict__ ln_b, const float* __restrict__ w1,
    const float* __restrict__ b1, const float* __restrict__ w2,
    const float* __restrict__ b2, float* __restrict__ out)
{
  // B operands pre-swizzled so each (kstep, ntile) WMMA needs one ds_load_b64:
  // entry[(k*NT + t)*32 + lane] = { W'[4k + 2*(lane>>4)    ][t*16 + (lane&15)],
  //                                 W'[4k + 2*(lane>>4) + 1][t*16 + (lane&15)] }
  __shared__ float2 sB1[K1 * N1 * 32];   // 38912 B
  __shared__ float2 sB2[K2 * N2 * 32];   // 16384 B
  __shared__ float2 sB3[K3 * N3 * 32];   //  8192 B
  __shared__ float  sc[NCONST];          //  2240 B
  __shared__ float  sx[4][16 * 64];      // 16384 B  per-wave D->A transpose buffer

  const int tid = threadIdx.x;

  // ---------------- stage weights into LDS (once per workgroup) ----------------
  for (int idx = tid; idx < K1 * N1 * 32; idx += 128) {
    int lane = idx & 31, t = (idx >> 5) & 3, k = idx >> 7;
    int col = t * 16 + (lane & 15);
    int r0 = 4 * k + 2 * (lane >> 4);
    float2 v;
    v.x = w1prime(r0,     col, chart_w, chart_b, tex_w);
    v.y = w1prime(r0 + 1, col, chart_w, chart_b, tex_w);
    sB1[idx] = v;
  }
  for (int idx = tid; idx < K2 * N2 * 32; idx += 128) {
    int lane = idx & 31, t = (idx >> 5) & 3, k = idx >> 7;
    int col = t * 16 + (lane & 15);
    int r0 = 4 * k + 2 * (lane >> 4);
    float2 v;                               // W2'[k][h] = w1[h][k]
    v.x = w1[col * 64 + r0];
    v.y = w1[col * 64 + r0 + 1];
    sB2[idx] = v;
  }
  for (int idx = tid; idx < K3 * N3 * 32; idx += 128) {
    int lane = idx & 31, t = (idx >> 5) & 1, k = idx >> 6;
    int col = t * 16 + (lane & 15);
    int r0 = 4 * k + 2 * (lane >> 4);
    float2 v;                               // W3'[k][o] = w2[o][k]
    v.x = w2[col * 64 + r0];
    v.y = w2[col * 64 + r0 + 1];
    sB3[idx] = v;
  }
  for (int i = tid; i < 128; i += 128) { sc[OFF_CEN + i] = centers[i]; sc[OFF_RWT + i] = router_w[i]; }
  for (int i = tid; i < 64; i += 128) {
    sc[OFF_TEXB + i] = tex_b[i]; sc[OFF_LNG + i] = ln_g[i];
    sc[OFF_LNB + i] = ln_b[i];   sc[OFF_B1 + i] = b1[i];
  }
  for (int i = tid; i < 32; i += 128) sc[OFF_B2 + i] = b2[i];
  for (int i = tid; i < 8; i += 128) {
    sc[OFF_RB + i] = router_b[i];
    float s = 0.f;
    for (int l = 0; l < 16; ++l) { float c = centers[i * 16 + l]; s += c * c; }
    sc[OFF_CSQ + i] = s;
  }
  __syncthreads();

  // ---------------- per-wave tile loop ----------------
  const int lane = tid & 31;
  const int wave = tid >> 5;
  const int m    = lane & 15;    // A-matrix row (batch row within tile)
  const int half = lane >> 4;    // K sub-offset selector
  const int gw = blockIdx.x * 4 + wave;
  const int nw = gridDim.x * 4;
  float* __restrict__ swx = sx[wave];
  const v8f zero8 = {0.f, 0.f, 0.f, 0.f, 0.f, 0.f, 0.f, 0.f};

  for (int tile = gw; tile < NTILES; tile += nw) {
    const int row = tile * 16 + m;
    const float* __restrict__ zg = z_geo + (size_t)row * 16;
    const float* __restrict__ zt = z_tex + (size_t)row * 16;

    // A-layout latents: zq[2*q+j] = z[4*q + 2*half + j]
    float zq[8], ztq[8];
#pragma unroll
    for (int q = 0; q < 4; ++q) {
      float2 p = *(const float2*)(zg + 4 * q + 2 * half);
      zq[2 * q] = p.x; zq[2 * q + 1] = p.y;
      float2 r = *(const float2*)(zt + 4 * q + 2 * half);
      ztq[2 * q] = r.x; ztq[2 * q + 1] = r.y;
    }
    {
      int nt = tile + nw;
      if (nt < NTILES) {
        __builtin_prefetch(z_geo + (size_t)(nt * 16 + m) * 16, 0, 0);
        __builtin_prefetch(z_tex + (size_t)(nt * 16 + m) * 16, 0, 0);
      }
    }

    // ---------------- router (per-lane, halves combined via shfl_xor 16) -------
    float p = 0.f;
#pragma unroll
    for (int i = 0; i < 8; ++i) p += zq[i] * zq[i];
    const float zz = p + __shfl_xor(p, 16, 32);

    float score[8];
#pragma unroll
    for (int c = 0; c < 8; ++c) {
      float dp = 0.f, rp = 0.f;
#pragma unroll
      for (int i = 0; i < 8; ++i) {
        int l = 4 * (i >> 1) + 2 * half + (i & 1);
        float cv = sc[OFF_CEN + c * 16 + l];
        float d = zq[i] - cv;
        dp += d * d;
        rp += zq[i] * sc[OFF_RWT + c * 16 + l];
      }
      float dsq  = dp + __shfl_xor(dp, 16, 32);
      float rdot = rp + __shfl_xor(rp, 16, 32) + sc[OFF_RB + c];
      float den  = (1.0f - zz) * (1.0f - sc[OFF_CSQ + c]) + 0.001f;
      float arg  = fmaxf(1.0f + 2.0f * dsq / den, 1.001f);
      score[c]   = -acoshf(arg) + 0.1f * rdot;
    }
    const float tau  = fmaxf(2.0f * fmaxf(1.0f - zz, 0.001f), 0.01f);
    const float itau = 1.0f / tau;
    float mx = -1e30f;
#pragma unroll
    for (int c = 0; c < 8; ++c) { score[c] *= itau; mx = fmaxf(mx, score[c]); }
    float rw[8]; float ssum = 0.f;
#pragma unroll
    for (int c = 0; c < 8; ++c) { rw[c] = expf(score[c] - mx); ssum += rw[c]; }
    const float isum = 1.0f / ssum;
#pragma unroll
    for (int c = 0; c < 8; ++c) rw[c] *= isum;
    // bias-blend A slots (K rows 128..135): rw[2*half + j], rw[4 + 2*half + j]
    const float ra0 = half ? rw[2] : rw[0];
    const float ra1 = half ? rw[3] : rw[1];
    const float ra2 = half ? rw[6] : rw[4];
    const float ra3 = half ? rw[7] : rw[5];

    // ---------------- GEMM1: h = A'[16x152] @ W1'[152x64] ----------------------
    v8f acc[4];
#pragma unroll
    for (int t = 0; t < 4; ++t) acc[t] = zero8;
#pragma unroll
    for (int k = 0; k < 32; ++k) {          // chart part, A = rw[c] * z
      int c = k >> 2, q = k & 3;
      v2f a; a.x = rw[c] * zq[2 * q]; a.y = rw[c] * zq[2 * q + 1];
#pragma unroll
      for (int t = 0; t < 4; ++t) {
        float2 bb = sB1[(k * 4 + t) * 32 + lane];
        v2f b; b.x = bb.x; b.y = bb.y;
        acc[t] = wmma4(a, b, acc[t]);
      }
    }
    {                                        // chart_b blend, A = rw
      v2f a; a.x = ra0; a.y = ra1;
#pragma unroll
      for (int t = 0; t < 4; ++t) {
        float2 bb = sB1[(32 * 4 + t) * 32 + lane];
        v2f b; b.x = bb.x; b.y = bb.y;
        acc[t] = wmma4(a, b, acc[t]);
      }
    }
    {
      v2f a; a.x = ra2; a.y = ra3;
#pragma unroll
      for (int t = 0; t < 4; ++t) {
        float2 bb = sB1[(33 * 4 + t) * 32 + lane];
        v2f b; b.x = bb.x; b.y = bb.y;
        acc[t] = wmma4(a, b, acc[t]);
      }
    }
#pragma unroll
    for (int k = 34; k < 38; ++k) {          // tex part, A = z_tex
      int q = k - 34;
      v2f a; a.x = ztq[2 * q]; a.y = ztq[2 * q + 1];
#pragma unroll
      for (int t = 0; t < 4; ++t) {
        float2 bb = sB1[(k * 4 + t) * 32 + lane];
        v2f b; b.x = bb.x; b.y = bb.y;
        acc[t] = wmma4(a, b, acc[t]);
      }
    }

    // + tex_b (column bias); D layout: lane&15 = column, vgpr v = row (v + 8*half)
#pragma unroll
    for (int t = 0; t < 4; ++t) {
      float tb = sc[OFF_TEXB + t * 16 + m];
#pragma unroll
      for (int v = 0; v < 8; ++v) acc[t][v] += tb;
    }

    // ---------------- LayerNorm over 64 cols per row ---------------------------
    float mu[8], rstd[8];
#pragma unroll
    for (int v = 0; v < 8; ++v) {
      float s  = acc[0][v] + acc[1][v] + acc[2][v] + acc[3][v];
      float q2 = acc[0][v] * acc[0][v] + acc[1][v] * acc[1][v] +
                 acc[2][v] * acc[2][v] + acc[3][v] * acc[3][v];
#pragma unroll
      for (int msk = 1; msk < 16; msk <<= 1) {
        s  += __shfl_xor(s,  msk, 32);
        q2 += __shfl_xor(q2, msk, 32);
      }
      float muv = s * (1.0f / 64.0f);
      float var = q2 * (1.0f / 64.0f) - muv * muv;
      mu[v] = muv;
      rstd[v] = rsqrtf(var + 1e-5f);
    }
    // normalize + GELU -> row-major LDS for D->A relayout
#pragma unroll
    for (int t = 0; t < 4; ++t) {
      int col = t * 16 + m;
      float g = sc[OFF_LNG + col], bo = sc[OFF_LNB + col];
#pragma unroll
      for (int v = 0; v < 8; ++v) {
        float x = (acc[t][v] - mu[v]) * rstd[v] * g + bo;
        swx[(v + 8 * half) * 64 + col] = gelu_exact(x);
      }
    }

    // ---------------- GEMM2: h1 = gelu(g @ w1^T + b1) --------------------------
    v8f a2[4];
#pragma unroll
    for (int t = 0; t < 4; ++t) a2[t] = zero8;
#pragma unroll
    for (int k = 0; k < K2; ++k) {
      float2 ap = *(const float2*)&swx[m * 64 + 4 * k + 2 * half];
      v2f a; a.x = ap.x; a.y = ap.y;
#pragma unroll
      for (int t = 0; t < 4; ++t) {
        float2 bb = sB2[(k * 4 + t) * 32 + lane];
        v2f b; b.x = bb.x; b.y = bb.y;
        a2[t] = wmma4(a, b, a2[t]);
      }
    }
#pragma unroll
    for (int t = 0; t < 4; ++t) {
      int col = t * 16 + m;
      float bo = sc[OFF_B1 + col];
#pragma unroll
      for (int v = 0; v < 8; ++v)
        swx[(v + 8 * half) * 64 + col] = gelu_exact(a2[t][v] + bo);
    }

    // ---------------- GEMM3: x_hat = h1 @ w2^T + b2 ----------------------------
    v8f a3[2];
#pragma unroll
    for (int t = 0; t < 2; ++t) a3[t] = zero8;
#pragma unroll
    for (int k = 0; k < K3; ++k) {
      float2 ap = *(const float2*)&swx[m * 64 + 4 * k + 2 * half];
      v2f a; a.x = ap.x; a.y = ap.y;
#pragma unroll
      for (int t = 0; t < 2; ++t) {
        float2 bb = sB3[(k * 2 + t) * 32 + lane];
        v2f b; b.x = bb.x; b.y = bb.y;
        a3[t] = wmma4(a, b, a3[t]);
      }
    }

    // ---------------- stores ---------------------------------------------------
    float* __restrict__ xo = out + (size_t)(tile * 16) * 32;
#pragma unroll
    for (int t = 0; t < 2; ++t) {
      float bo = sc[OFF_B2 + t * 16 + m];
#pragma unroll
      for (int v = 0; v < 8; ++v)
        xo[(v + 8 * half) * 32 + t * 16 + m] = a3[t][v] + bo;
    }
    if (half == 0) {   // router_weights, one lane per row
      float* __restrict__ ro = out + (size_t)BATCH * 32 + (size_t)row * 8;
      float4 r0; r0.x = rw[0]; r0.y = rw[1]; r0.z = rw[2]; r0.w = rw[3];
      float4 r1; r1.x = rw[4]; r1.y = rw[5]; r1.z = rw[6]; r1.w = rw[7];
      *(float4*)(ro)     = r0;
      *(float4*)(ro + 4) = r1;
    }
  }
}

extern "C" void kernel_launch(void* const* d_in, const int* in_sizes, int n_in,
                              void* d_out, int out_size, void* d_ws, size_t ws_size,
                              hipStream_t stream) {
  const float* z_geo    = (const float*)d_in[0];
  const float* z_tex    = (const float*)d_in[1];
  const float* chart_w  = (const float*)d_in[2];
  const float* chart_b  = (const float*)d_in[3];
  const float* router_w = (const float*)d_in[4];
  const float* router_b = (const float*)d_in[5];
  const float* centers  = (const float*)d_in[6];
  const float* tex_w    = (const float*)d_in[7];
  const float* tex_b    = (const float*)d_in[8];
  const float* ln_g     = (const float*)d_in[9];
  const float* ln_b     = (const float*)d_in[10];
  const float* w1       = (const float*)d_in[11];
  const float* b1       = (const float*)d_in[12];
  const float* w2       = (const float*)d_in[13];
  const float* b2       = (const float*)d_in[14];

  dim3 grid(1024), block(128);
  topo_decoder_kernel<<<grid, block, 0, stream>>>(
      z_geo, z_tex, chart_w, chart_b, router_w, router_b, centers,
      tex_w, tex_b, ln_g, ln_b, w1, b1, w2, b2, (float*)d_out);
}